// PyQObservable_24043226923475
// MI455X (gfx1250) — compile-verified
//
#include <hip/hip_runtime.h>

// ---------------------------------------------------------------------------
// out[t,r,b] = sum_j M[b][t][j] * state[j,r,b]  (complex), state = (32, 2^15, 16)
// M[b] = sum_k I (x) U_k[b] (x) I on the top-5-bit space, U_k = exp(-i H_k t).
// Kernel2: per-batch complex GEMM on V_WMMA_F32_16X16X4_F32, LDS-staged I/O.
// ---------------------------------------------------------------------------
#define BATCH   16
#define NTERMS  4
#define TDIM    32
#define RDIM    32768        // 2^15
#define DIMTOT  1048576      // 2^20

#define RSTRIDE 20                    // dwords per r-row in LDS (16 b + 4 pad)
#define TSTRIDE (RSTRIDE * 16)        // 320 dwords per t-slab
#define PLANE_DW (TDIM * TSTRIDE)     // 10240 dwords per plane (40 KB)
#define SMEM_BYTES (2 * PLANE_DW * 4) // 80 KB

typedef float v2f __attribute__((ext_vector_type(2)));
typedef float v4f __attribute__((ext_vector_type(4)));
typedef float v8f __attribute__((ext_vector_type(8)));

// ---------------- small complex 4x4 helpers (kernel 1 only) ----------------
__device__ inline void cmm4(const float ar[4][4], const float ai[4][4],
                            const float br[4][4], const float bi[4][4],
                            float cr[4][4], float ci[4][4]) {
#pragma unroll
  for (int i = 0; i < 4; ++i)
#pragma unroll
    for (int j = 0; j < 4; ++j) {
      float sr = 0.f, si = 0.f;
#pragma unroll
      for (int l = 0; l < 4; ++l) {
        sr += ar[i][l] * br[l][j] - ai[i][l] * bi[l][j];
        si += ar[i][l] * bi[l][j] + ai[i][l] * br[l][j];
      }
      cr[i][j] = sr; ci[i][j] = si;
    }
}

__device__ inline void cexpm4(float gre[4][4], float gim[4][4],
                              float ere[4][4], float eim[4][4]) {
  float nrm = 0.f;
#pragma unroll
  for (int i = 0; i < 4; ++i) {
    float rs = 0.f;
#pragma unroll
    for (int j = 0; j < 4; ++j) rs += fabsf(gre[i][j]) + fabsf(gim[i][j]);
    nrm = fmaxf(nrm, rs);
  }
  int s = 0;
  while (nrm > 0.5f && s < 30) { nrm *= 0.5f; ++s; }
  const float sc = ldexpf(1.f, -s);
#pragma unroll
  for (int i = 0; i < 4; ++i)
#pragma unroll
    for (int j = 0; j < 4; ++j) { gre[i][j] *= sc; gim[i][j] *= sc; }

  float tre[4][4], tim[4][4], rre[4][4], rim[4][4];
#pragma unroll
  for (int i = 0; i < 4; ++i)
#pragma unroll
    for (int j = 0; j < 4; ++j) {
      float id = (i == j) ? 1.f : 0.f;
      ere[i][j] = id; eim[i][j] = 0.f;
      tre[i][j] = id; tim[i][j] = 0.f;
    }
  for (int n = 1; n <= 12; ++n) {           // Taylor: T <- T*G/n ; E += T
    cmm4(tre, tim, gre, gim, rre, rim);
    const float inv = 1.f / (float)n;
#pragma unroll
    for (int i = 0; i < 4; ++i)
#pragma unroll
      for (int j = 0; j < 4; ++j) {
        tre[i][j] = rre[i][j] * inv; tim[i][j] = rim[i][j] * inv;
        ere[i][j] += tre[i][j];      eim[i][j] += tim[i][j];
      }
  }
  for (int q = 0; q < s; ++q) {             // repeated squaring
    cmm4(ere, eim, ere, eim, rre, rim);
#pragma unroll
    for (int i = 0; i < 4; ++i)
#pragma unroll
      for (int j = 0; j < 4; ++j) { ere[i][j] = rre[i][j]; eim[i][j] = rim[i][j]; }
  }
}

// ---------------------------------------------------------------------------
// Kernel 1: assemble M[b] (32x32 complex, re/im interleaved) into d_ws.
// ---------------------------------------------------------------------------
__global__ void build_M_kernel(const float* __restrict__ Hr,
                               const float* __restrict__ Hi,
                               const float* __restrict__ tevo,
                               float* __restrict__ Mbuf) {
  const int b = threadIdx.x;
  if (b >= BATCH) return;

  float Ure[NTERMS][4][4], Uim[NTERMS][4][4];
  for (int k = 0; k < NTERMS; ++k) {
    float hre[4][4], him[4][4];
#pragma unroll
    for (int i = 0; i < 4; ++i)
#pragma unroll
      for (int j = 0; j < 4; ++j) {
        hre[i][j] = 0.5f * (Hr[k * 16 + i * 4 + j] + Hr[k * 16 + j * 4 + i]);
        him[i][j] = 0.5f * (Hi[k * 16 + i * 4 + j] - Hi[k * 16 + j * 4 + i]);
      }
    const float t = tevo[k * BATCH + b];
    float gre[4][4], gim[4][4];            // G = -i*H*t
#pragma unroll
    for (int i = 0; i < 4; ++i)
#pragma unroll
      for (int j = 0; j < 4; ++j) { gre[i][j] = him[i][j] * t; gim[i][j] = -hre[i][j] * t; }
    cexpm4(gre, gim, Ure[k], Uim[k]);
  }

  for (int i = 0; i < TDIM; ++i)
    for (int j = 0; j < TDIM; ++j) {
      float mr = 0.f, mi = 0.f;
#pragma unroll
      for (int k = 0; k < NTERMS; ++k) {
        const int sh = 3 - k;
        const int msk = 31 & ~(3 << sh);
        if ((i & msk) == (j & msk)) {
          mr += Ure[k][(i >> sh) & 3][(j >> sh) & 3];
          mi += Uim[k][(i >> sh) & 3][(j >> sh) & 3];
        }
      }
      const long o = (((long)b * TDIM + i) * TDIM + j) * 2;
      Mbuf[o] = mr; Mbuf[o + 1] = mi;
    }
}

// ---------------------------------------------------------------------------
// Kernel 2: LDS-staged batched complex GEMM.
//   Block = 16 waves; wave w owns batch b=w.  r-tile = 16.
//   fill  : coalesced global_load_b128 (NT) -> ds_store_b128 (padded layout)
//   math  : B frags via ds_load_b32 (bank-spread), A frags 16B from M[b],
//           64 x v_wmma_f32_16x16x4_f32 per wave
//   drain : acc frags -> LDS (reused) -> ds_load_b128 -> global_store_b128 (NT)
// ---------------------------------------------------------------------------
__global__ __launch_bounds__(512) void apply_gates_kernel(
    const float* __restrict__ sre, const float* __restrict__ sim,
    const float* __restrict__ Mbuf, float* __restrict__ out) {
  extern __shared__ float smem[];                 // [2][PLANE_DW]
  float* Lp[2] = { smem, smem + PLANE_DW };

  const int tid  = threadIdx.x;
  const int lane = tid & 31;
  const int b    = tid >> 5;                      // wave id == batch index
  const int nl   = lane & 15;
  const int kh   = lane >> 4;
  const long r0  = (long)blockIdx.x * 16;
  const long gtile = r0 * BATCH;                  // float offset of (t=0, r0, b=0)

  const float* gp[2] = { sre, sim };

  // ---- fill: 4 rounds x v4f per plane; L = linear float idx in 32x16x16 tile
#pragma unroll
  for (int p = 0; p < 2; ++p)
#pragma unroll
    for (int i = 0; i < 4; ++i) {
      const int L = i * 2048 + tid * 4;
      const int t = L >> 8, w = L & 255;          // w = r*16 + b0 (b0 % 4 == 0)
      const int r = w >> 4, b0 = w & 15;
      const v4f d = __builtin_nontemporal_load(
          (const v4f*)(gp[p] + ((long)t * RDIM) * BATCH + gtile + w));
      *(v4f*)(Lp[p] + t * TSTRIDE + r * RSTRIDE + b0) = d;
    }
  __syncthreads();

  // ---- math
  v8f accR0 = {}, accR1 = {}, accI0 = {}, accI1 = {};
#pragma unroll
  for (int kk = 0; kk < 8; ++kk) {
    const int kbase = kk * 4 + 2 * kh;            // lane-half K offset
    const int i0 = kbase * TSTRIDE + nl * RSTRIDE + b;
    v2f Bre, Bim;
    Bre.x = Lp[0][i0];            Bre.y = Lp[0][i0 + TSTRIDE];
    Bim.x = Lp[1][i0];            Bim.y = Lp[1][i0 + TSTRIDE];

#pragma unroll
    for (int m = 0; m < 2; ++m) {
      const int row = m * 16 + nl;
      const v4f a4 = *(const v4f*)(Mbuf + ((((long)b * TDIM + row) * TDIM + kbase) * 2));
      v2f Are, Aim, AimN;
      Are.x = a4.x; Aim.x = a4.y;
      Are.y = a4.z; Aim.y = a4.w;
      AimN  = -Aim;                               // f32 WMMA has no A-neg modifier

      v8f aR = (m == 0) ? accR0 : accR1;
      v8f aI = (m == 0) ? accI0 : accI1;
      aR = __builtin_amdgcn_wmma_f32_16x16x4_f32(false, Are,  false, Bre, (short)0, aR, false, false);
      aR = __builtin_amdgcn_wmma_f32_16x16x4_f32(false, AimN, false, Bim, (short)0, aR, false, false);
      aI = __builtin_amdgcn_wmma_f32_16x16x4_f32(false, Are,  false, Bim, (short)0, aI, false, false);
      aI = __builtin_amdgcn_wmma_f32_16x16x4_f32(false, Aim,  false, Bre, (short)0, aI, false, false);
      if (m == 0) { accR0 = aR; accI0 = aI; } else { accR1 = aR; accI1 = aI; }
    }
  }
  __syncthreads();                                // frag reads done; reuse LDS

  // ---- scatter accumulators into LDS (C/D layout: lanes16-31 are M+8)
#pragma unroll
  for (int m = 0; m < 2; ++m) {
    const v8f aR = (m == 0) ? accR0 : accR1;
    const v8f aI = (m == 0) ? accI0 : accI1;
#pragma unroll
    for (int v = 0; v < 8; ++v) {
      const int t = m * 16 + v + kh * 8;
      const int o = t * TSTRIDE + nl * RSTRIDE + b;
      Lp[0][o] = aR[v];
      Lp[1][o] = aI[v];
    }
  }
  __syncthreads();

  // ---- drain: coalesced stores, same linearization as fill
  const long PLANE = (long)DIMTOT * BATCH;
#pragma unroll
  for (int p = 0; p < 2; ++p)
#pragma unroll
    for (int i = 0; i < 4; ++i) {
      const int L = i * 2048 + tid * 4;
      const int t = L >> 8, w = L & 255;
      const int r = w >> 4, b0 = w & 15;
      const v4f d = *(const v4f*)(Lp[p] + t * TSTRIDE + r * RSTRIDE + b0);
      __builtin_nontemporal_store(
          d, (v4f*)(out + p * PLANE + ((long)t * RDIM) * BATCH + gtile + w));
    }
}

// ---------------------------------------------------------------------------
extern "C" void kernel_launch(void* const* d_in, const int* in_sizes, int n_in,
                              void* d_out, int out_size, void* d_ws, size_t ws_size,
                              hipStream_t stream) {
  const float* Hr   = (const float*)d_in[0];  // (4,4,4)
  const float* Hi   = (const float*)d_in[1];  // (4,4,4)
  const float* tevo = (const float*)d_in[2];  // (4,16)
  const float* sre  = (const float*)d_in[3];  // (DIM,16)
  const float* sim  = (const float*)d_in[4];  // (DIM,16)
  float* Mbuf = (float*)d_ws;                 // 16*32*32*2 floats = 128 KB
  float* out  = (float*)d_out;                // (2, DIM, 16)

  build_M_kernel<<<dim3(1), dim3(32), 0, stream>>>(Hr, Hi, tevo, Mbuf);
  apply_gates_kernel<<<dim3(RDIM / 16), dim3(512), SMEM_BYTES, stream>>>(sre, sim, Mbuf, out);
}